// DilatedKnnGraph_89816356094631
// MI455X (gfx1250) — compile-verified
//
#include <hip/hip_runtime.h>

typedef __attribute__((ext_vector_type(2))) float v2f;
typedef __attribute__((ext_vector_type(4))) float v4f;
typedef __attribute__((ext_vector_type(8))) float v8f;

#define KNN   9
#define KD    18      // K * DILATION candidates
#define NDIM  128
#define NPTS  4096
#define NCLD  8
#define CSTR  132     // padded LDS row stride (floats) -> conflict-free b64 reads

// ---------------------------------------------------------------------------
// Kernel 1: per-point squared norms |x_j|^2
// ---------------------------------------------------------------------------
__global__ void sqnorm_kernel(const float* __restrict__ x,
                              float* __restrict__ sq, int ntot) {
    int i = blockIdx.x * blockDim.x + threadIdx.x;
    if (i >= ntot) return;
    const v4f* p = (const v4f*)(x + (size_t)i * NDIM);
    float acc = 0.f;
#pragma unroll
    for (int j = 0; j < NDIM / 4; ++j) {
        v4f v = p[j];
        acc += v.x * v.x + v.y * v.y + v.z * v.z + v.w * v.w;
    }
    sq[i] = acc;
}

// ---------------------------------------------------------------------------
// Kernel 2: one wave32 per 16-row tile. Double-buffered async LDS staging,
// fp32 WMMA inner products, top-18 selection, dilated edge emission.
// ---------------------------------------------------------------------------
__global__ __launch_bounds__(32) void knn_kernel(const float* __restrict__ x,
                                                 const float* __restrict__ sq,
                                                 int* __restrict__ out) {
    __shared__ __align__(16) float colbuf[2][16 * CSTR]; // double-buffered tile
    __shared__ float distT[16 * 17];                     // transposed dist tile
    __shared__ float mergeD[32 * KD];
    __shared__ int   mergeI[32 * KD];

    const int lane = threadIdx.x;        // 0..31
    const int h    = lane >> 4;          // lane-half (ISA K/row-half select)
    const int l16  = lane & 15;
    const int tile = blockIdx.x;         // 0..2047
    const int cloud = tile / (NPTS / 16);
    const int r0    = (tile % (NPTS / 16)) * 16;

    const float* xc  = x  + (size_t)cloud * NPTS * NDIM;
    const float* sqc = sq + cloud * NPTS;
    const unsigned long long xbase = (unsigned long long)(const void*)xc;
    const unsigned ldsbase0 = (unsigned)(size_t)(void*)&colbuf[0][0];
    const unsigned ldsbase1 = (unsigned)(size_t)(void*)&colbuf[1][0];

    // --- A fragments: 16x4 fp32 layout -> lane l16 = row M, half selects K pair.
    v2f afrag[32];
    const float* arow = xc + (size_t)(r0 + l16) * NDIM;
#pragma unroll
    for (int kc = 0; kc < 32; ++kc)
        afrag[kc] = *(const v2f*)(arow + kc * 4 + 2 * h);

    // --- per-lane sorted candidate list (lane owns row l16, column-half h)
    float bestD[KD];
    int   bestI[KD];
#pragma unroll
    for (int t = 0; t < KD; ++t) { bestD[t] = 3.4e38f; bestI[t] = 0; }

    // Async-stage one 16-point column tile: 16 ops, each moves one point row
    // (32 lanes x 16B = 512B = 128 floats), GVS addressing, ASYNCcnt-tracked.
    auto stage = [&](int j0, unsigned ldsb) {
#pragma unroll
        for (int i = 0; i < 16; ++i) {
            unsigned voff = (unsigned)((j0 + i) * (NDIM * 4)) + lane * 16;
            unsigned doff = ldsb + i * (CSTR * 4) + lane * 16;
            asm volatile("global_load_async_to_lds_b128 %0, %1, %2"
                         :
                         : "v"(doff), "v"(voff), "s"(xbase)
                         : "memory");
        }
    };

    stage(0, ldsbase0);                                  // prologue

    for (int t = 0; t < NPTS / 16; ++t) {
        const int j0 = t * 16;
        const unsigned curb = (t & 1) ? ldsbase1 : ldsbase0;
        const float* cur = (t & 1) ? &colbuf[1][0] : &colbuf[0][0];

        if (t + 1 < NPTS / 16) {                         // prefetch next tile
            stage(j0 + 16, (t & 1) ? ldsbase0 : ldsbase1);
            asm volatile("s_wait_asynccnt 0x10" ::: "memory"); // gate current 16
        } else {
            asm volatile("s_wait_asynccnt 0x0" ::: "memory");
        }

        // B fragment: lane l16 = column N, half selects K pair.
        v8f acc = {};
#pragma unroll
        for (int kc = 0; kc < 32; ++kc) {
            v2f b = *(const v2f*)(cur + l16 * CSTR + kc * 4 + 2 * h);
            acc = __builtin_amdgcn_wmma_f32_16x16x4_f32(
                false, afrag[kc], false, b, (short)0, acc, false, false);
        }

        // C layout: lane holds column N=l16, rows M = v + 8*h.
        // Order-equivalent distance: |x_j|^2 - 2 * <x_i, x_j>
        const float sqj = sqc[j0 + l16];
#pragma unroll
        for (int v = 0; v < 8; ++v) {
            int m = v + 8 * h;
            distT[m * 17 + l16] = sqj - 2.0f * acc[v];
        }
        __syncthreads();

        // selection: lane owns row l16, columns h*8 .. h*8+7 of this tile
        float worst = bestD[KD - 1];
#pragma unroll
        for (int q = 0; q < 8; ++q) {
            int   c   = h * 8 + q;
            float d   = distT[l16 * 17 + c];
            int   idx = j0 + c;
            if (d < worst) {
                float cd = d; int ci = idx;
#pragma unroll
                for (int u = 0; u < KD; ++u) {   // branch-free sorted insert
                    float td = bestD[u]; int ti = bestI[u];
                    bool  lt = cd < td;
                    bestD[u] = lt ? cd : td;
                    bestI[u] = lt ? ci : ti;
                    cd = lt ? td : cd;
                    ci = lt ? ti : ci;
                }
                worst = bestD[KD - 1];
            }
        }
        __syncthreads();  // protect distT before next tile
    }

    // --- merge the two per-row half-lists, emit every 2nd of first 18
#pragma unroll
    for (int t = 0; t < KD; ++t) {
        mergeD[lane * KD + t] = bestD[t];
        mergeI[lane * KD + t] = bestI[t];
    }
    __syncthreads();

    if (lane < 16) {
        const int g = cloud * NPTS + r0 + lane;        // center global index
        int* out0 = out;                                // neighbor ids
        int* out1 = out + NCLD * NPTS * KNN;            // center ids
        int p = 0, q = 0;
#pragma unroll
        for (int t = 0; t < KD; ++t) {                  // p+q = t <= 17: in-range
            float da = mergeD[lane * KD + p];
            float db = mergeD[(lane + 16) * KD + q];
            int   ia = mergeI[lane * KD + p];
            int   ib = mergeI[(lane + 16) * KD + q];
            bool  ta = (da < db) || (da == db && ia < ib);
            int  sel = ta ? ia : ib;
            if (ta) ++p; else ++q;
            if ((t & 1) == 0) {                         // dilated [::2][:9]
                out0[(size_t)g * KNN + (t >> 1)] = cloud * NPTS + sel;
                out1[(size_t)g * KNN + (t >> 1)] = g;
            }
        }
    }
}

// ---------------------------------------------------------------------------
extern "C" void kernel_launch(void* const* d_in, const int* in_sizes, int n_in,
                              void* d_out, int out_size, void* d_ws, size_t ws_size,
                              hipStream_t stream) {
    const float* x  = (const float*)d_in[0];
    float*       sq = (float*)d_ws;          // 32768 floats = 128 KB scratch
    int*         out = (int*)d_out;
    int ntot = in_sizes[0] / NDIM;           // 32768 points

    sqnorm_kernel<<<(ntot + 255) / 256, 256, 0, stream>>>(x, sq, ntot);
    knn_kernel<<<ntot / 16, 32, 0, stream>>>(x, sq, out);
}